// Indexer_24515673325873
// MI455X (gfx1250) — compile-verified
//
#include <hip/hip_runtime.h>
#include <hip/hip_bf16.h>

// Gather kernel for: out[i,:] = items[min(floor(clip(ind[i],0,1)*n_items), n_items-1), :]
// N = 2^20 rows, D = 256 f32 (64 float4 chunks per row).
//
// Bandwidth-bound: ~1 GiB streamed output. Strategy:
//  - async-stage each block's 256 index values into LDS (CDNA5
//    GLOBAL_LOAD_ASYNC_TO_LDS_B32, ASYNCcnt + s_wait_asynccnt)
//  - convert once to row ids in LDS
//  - 64 threads/row copy float4 chunks: cached b128 loads from the 1 MiB
//    items table (L2-resident), non-temporal b128 stores for the output
//    stream so it doesn't evict items from L2.

#define TILE   256   // indices (output rows) per block
#define D_ITEM 256   // floats per row
#define D4     (D_ITEM / 4)   // float4 chunks per row = 64

typedef float v4f __attribute__((ext_vector_type(4)));

__global__ __launch_bounds__(TILE) void Indexer_gather_kernel(
    const float* __restrict__ indices,
    const v4f*   __restrict__ items4,
    v4f*         __restrict__ out4,
    int n, int n_items)
{
    __shared__ float sIdx[TILE];
    __shared__ int   sRow[TILE];

    const int t        = threadIdx.x;
    const int tileBase = blockIdx.x * TILE;

    // ---- Stage index tile into LDS via CDNA5 async global->LDS copy ----
    {
        int src = tileBase + t;
        if (src >= n) src = n - 1;              // keep all lanes active (EXEC all 1s)
        const float* gp = indices + src;
        // Low 32 bits of a flat shared pointer == LDS byte offset (ISA 10.2).
        unsigned ldsAddr = (unsigned)(size_t)(&sIdx[t]);
        asm volatile("global_load_async_to_lds_b32 %0, %1, off"
                     :: "v"(ldsAddr), "v"(gp)
                     : "memory");
        asm volatile("s_wait_asynccnt 0" ::: "memory");
    }
    __syncthreads();

    // ---- Convert to clamped row ids (one thread per index) ----
    {
        float x = sIdx[t];
        x = fminf(fmaxf(x, 0.0f), 1.0f);
        int r = (int)(x * (float)n_items);      // x >= 0, trunc == floor
        if (r > n_items - 1) r = n_items - 1;
        sRow[t] = r;
    }
    __syncthreads();

    // ---- Copy loop: 64 threads per row, one float4 each per row ----
    const int chunk = t & (D4 - 1);   // 0..63, column chunk
    const int rbase = t >> 6;         // 0..3, which row within the 4-row slab

    #pragma unroll 8
    for (int i = 0; i < TILE / 4; ++i) {
        int rl   = i * 4 + rbase;               // local row in tile (0..255)
        int grow = tileBase + rl;               // global output row
        if (grow < n) {
            int r = sRow[rl];                   // ds_load broadcast (uniform per 64 lanes)
            v4f v = items4[(size_t)r * D4 + chunk];             // cached b128 load
            __builtin_nontemporal_store(v, &out4[(size_t)grow * D4 + chunk]); // NT b128 store
        }
    }
}

extern "C" void kernel_launch(void* const* d_in, const int* in_sizes, int n_in,
                              void* d_out, int out_size, void* d_ws, size_t ws_size,
                              hipStream_t stream) {
    const float* indices = (const float*)d_in[0];
    const float* items   = (const float*)d_in[1];
    float*       out     = (float*)d_out;

    const int n       = in_sizes[0];            // number of query indices
    const int n_items = in_sizes[1] / D_ITEM;   // rows in items table

    const int blocks = (n + TILE - 1) / TILE;
    Indexer_gather_kernel<<<blocks, TILE, 0, stream>>>(
        indices, (const v4f*)items, (v4f*)out, n, n_items);
}